// CapLayer_68178310857152
// MI455X (gfx1250) — compile-verified
//
#include <hip/hip_runtime.h>

typedef float v2f __attribute__((ext_vector_type(2)));
typedef float v8f __attribute__((ext_vector_type(8)));

#define BS 64
#define NUM_SHARED 32
#define IN_DIM 32
#define OUT_DIM 64
#define HW 256          // H*W_SP = 16*16
#define NUM_IN 8192     // NUM_SHARED * HW
#define ROUTE_NUM 3
#define NSPLIT 4        // workgroups per batch for the routing pass

static __device__ __forceinline__ v8f wmma4(v2f a, v2f b, v8f c) {
  return __builtin_amdgcn_wmma_f32_16x16x4_f32(false, a, false, b, (short)0, c, false, false);
}

// ---------------------------------------------------------------------------
// Kernel A: pred[b, s*256+p, o] = sum_i u[b,s,p,i] * W[s,o,i]
// One workgroup per (s,b); 8 waves, each wave owns 2 row-tiles of 16.
// ---------------------------------------------------------------------------
__global__ __launch_bounds__(256)
void pred_gemm_kernel(const float* __restrict__ x,
                      const float* __restrict__ W,
                      float* __restrict__ pred) {
  const int s    = blockIdx.x;
  const int b    = blockIdx.y;
  const int lane = threadIdx.x & 31;
  const int wv   = threadIdx.x >> 5;
  const int lo16 = lane & 15;
  const int hi   = lane >> 4;

  const float* u  = x + (size_t)(b * NUM_SHARED + s) * HW * IN_DIM;   // [256][32]
  const float* Ws = W + (size_t)s * OUT_DIM * IN_DIM;                 // [64][32]
  float* P = pred + ((size_t)b * NUM_IN + (size_t)s * HW) * OUT_DIM;  // [256][64]

  // B fragments: B[k=i][n=o] = Ws[o][i];  4 o-tiles x 8 k-chunks
  v2f Bf[4][8];
#pragma unroll
  for (int t = 0; t < 4; ++t) {
    const int o = t * 16 + lo16;
#pragma unroll
    for (int kk = 0; kk < 8; ++kk)
      Bf[t][kk] = *(const v2f*)(Ws + o * IN_DIM + kk * 4 + 2 * hi);
  }

#pragma unroll
  for (int mm = 0; mm < 2; ++mm) {
    const int p0 = (wv * 2 + mm) * 16;
    v2f Af[8];
#pragma unroll
    for (int kk = 0; kk < 8; ++kk)
      Af[kk] = *(const v2f*)(u + (p0 + lo16) * IN_DIM + kk * 4 + 2 * hi);

    v8f D[4] = {};
#pragma unroll
    for (int kk = 0; kk < 8; ++kk)
#pragma unroll
      for (int t = 0; t < 4; ++t)
        D[t] = wmma4(Af[kk], Bf[t][kk], D[t]);

#pragma unroll
    for (int t = 0; t < 4; ++t)
#pragma unroll
      for (int r = 0; r < 8; ++r)
        P[(size_t)(p0 + r + 8 * hi) * OUT_DIM + t * 16 + lo16] = D[t][r];
  }
}

// ---------------------------------------------------------------------------
// Zero the running Vsum accumulator (global, per batch).
// ---------------------------------------------------------------------------
__global__ __launch_bounds__(256)
void zero_v_kernel(float* __restrict__ Vg) {
  Vg[(size_t)blockIdx.x * 4096 + threadIdx.x * 16 + 0] = 0.0f;
  for (int i = 1; i < 16; ++i)
    Vg[(size_t)blockIdx.x * 4096 + threadIdx.x * 16 + i] = 0.0f;
}

// ---------------------------------------------------------------------------
// Kernel B: one routing pass (one iteration), split NSPLIT ways per batch.
//   logits[o,n] = pred[n,:]·Vsum[o,:]   (b never materialized: b_t = pred@Vsumᵀ)
//   c = softmax_o(logits);  Spart += cᵀ-weighted pred  (deterministic reduce)
// grid = (NSPLIT, BS), block = 256 (8 waves); each wave owns 256 n's.
// ---------------------------------------------------------------------------
__global__ __launch_bounds__(256)
void route_pass_kernel(const float* __restrict__ pred,
                       const float* __restrict__ Vg,
                       float* __restrict__ Spart) {
  __shared__ float Vsum[64 * 66];            // padded stride 66 (bank spread, 8B align)
  __shared__ float shmem[8 * 64 * 18];       // union: per-wave c-bounce / reduce staging
  const int sp   = blockIdx.x;
  const int b    = blockIdx.y;
  const int tid  = threadIdx.x;
  const int lane = tid & 31;
  const int wv   = tid >> 5;
  const int lo16 = lane & 15;
  const int hi   = lane >> 4;

  const float* P = pred + (size_t)b * NUM_IN * OUT_DIM;
  float* cw = shmem + wv * (64 * 18);        // per-wave c buffer, stride 18

  // stage Vsum into LDS (padded)
  for (int i = tid; i < 4096; i += 256)
    Vsum[(i >> 6) * 66 + (i & 63)] = Vg[(size_t)b * 4096 + i];
  __syncthreads();

  v8f acc[4][4] = {};                        // per-wave partial S (64x64)
  const int n0base = (sp * 8 + wv) * 256;    // this wave's 256-row slice of n

  for (int nt = 0; nt < 16; ++nt) {
    const int n0 = n0base + nt * 16;

    // ---- GEMM1: logits tile [o=64][n=16] = Vsum(A) x predᵀ(B), K=64 ----
    v8f Dl[4] = {};
    const float* Prow = P + (size_t)(n0 + lo16) * OUT_DIM;
#pragma unroll
    for (int kk = 0; kk < 16; ++kk) {
      const v2f bfr = *(const v2f*)(Prow + kk * 4 + 2 * hi);
#pragma unroll
      for (int m = 0; m < 4; ++m) {
        const v2f afr = *(const v2f*)(&Vsum[(m * 16 + lo16) * 66 + kk * 4 + 2 * hi]);
        Dl[m] = wmma4(afr, bfr, Dl[m]);
      }
    }

    // ---- softmax over o: 8 regs x 4 tiles per lane, + lane^16 half ----
    float mx = -3.402823466e38f;
#pragma unroll
    for (int m = 0; m < 4; ++m)
#pragma unroll
      for (int r = 0; r < 8; ++r) mx = fmaxf(mx, Dl[m][r]);
    mx = fmaxf(mx, __shfl_xor(mx, 16, 32));
    float sum = 0.0f;
#pragma unroll
    for (int m = 0; m < 4; ++m)
#pragma unroll
      for (int r = 0; r < 8; ++r) {
        const float e = __expf(Dl[m][r] - mx);
        Dl[m][r] = e;
        sum += e;
      }
    sum += __shfl_xor(sum, 16, 32);
    const float inv = 1.0f / sum;

    // ---- bounce c to LDS (D-layout -> A-layout transpose) ----
#pragma unroll
    for (int m = 0; m < 4; ++m)
#pragma unroll
      for (int r = 0; r < 8; ++r)
        cw[(m * 16 + r + 8 * hi) * 18 + lo16] = Dl[m][r] * inv;
    asm volatile("s_wait_dscnt 0x0" ::: "memory");   // intra-wave LDS RAW fence

    // ---- GEMM2: acc[o,d] += c(A, K=16) x pred(B) ----
#pragma unroll
    for (int kk = 0; kk < 4; ++kk) {
      v2f af[4];
#pragma unroll
      for (int m = 0; m < 4; ++m)
        af[m] = *(const v2f*)(&cw[(m * 16 + lo16) * 18 + kk * 4 + 2 * hi]);
#pragma unroll
      for (int dt = 0; dt < 4; ++dt) {
        v2f bf;
        bf.x = P[(size_t)(n0 + kk * 4 + 2 * hi) * OUT_DIM + dt * 16 + lo16];
        bf.y = P[(size_t)(n0 + kk * 4 + 2 * hi + 1) * OUT_DIM + dt * 16 + lo16];
#pragma unroll
        for (int m = 0; m < 4; ++m)
          acc[m][dt] = wmma4(af[m], bf, acc[m][dt]);
      }
    }
  } // nt

  // ---- deterministic cross-wave reduction, 16-row chunks through LDS ----
  float* Sout = Spart + (size_t)(b * NSPLIT + sp) * 4096;
  __syncthreads();                            // c-bounce region now reusable
#pragma unroll 1
  for (int m = 0; m < 4; ++m) {
#pragma unroll
    for (int dt = 0; dt < 4; ++dt)
#pragma unroll
      for (int r = 0; r < 8; ++r)
        shmem[wv * 1024 + (r + 8 * hi) * 64 + dt * 16 + lo16] = acc[m][dt][r];
    __syncthreads();
    for (int e = tid; e < 1024; e += 256) {
      float t = shmem[e];                     // fixed wave order -> deterministic
      for (int w = 1; w < 8; ++w) t += shmem[w * 1024 + e];
      Sout[m * 1024 + e] = t;
    }
    __syncthreads();
  }
}

// ---------------------------------------------------------------------------
// Kernel C: squash + Vsum update (+ output on last iteration).
// grid = BS, block = 64 (thread == out-capsule o).
// ---------------------------------------------------------------------------
__global__ __launch_bounds__(64)
void squash_kernel(const float* __restrict__ Spart,
                   float* __restrict__ Vg,
                   float* __restrict__ vout,
                   int write_out) {
  const int b = blockIdx.x;
  const int o = threadIdx.x;
  float sv[64];
  float nsq = 0.0f;
#pragma unroll
  for (int d = 0; d < 64; ++d) {
    float t = Spart[(size_t)(b * NSPLIT + 0) * 4096 + o * 64 + d];
#pragma unroll
    for (int sp = 1; sp < NSPLIT; ++sp)
      t += Spart[(size_t)(b * NSPLIT + sp) * 4096 + o * 64 + d];
    sv[d] = t;
    nsq += t * t;
  }
  const float coeff = sqrtf(nsq) / (1.0f + nsq);  // |s|/(1+|s|²) == |s|²/(1+|s|²)/|s|
#pragma unroll
  for (int d = 0; d < 64; ++d) {
    const float vd = sv[d] * coeff;
    Vg[(size_t)b * 4096 + o * 64 + d] += vd;
    if (write_out) vout[((size_t)b * 64 + o) * 64 + d] = vd;
  }
}

extern "C" void kernel_launch(void* const* d_in, const int* in_sizes, int n_in,
                              void* d_out, int out_size, void* d_ws, size_t ws_size,
                              hipStream_t stream) {
  const float* x = (const float*)d_in[0];   // [64, 1024, 16, 16]
  const float* W = (const float*)d_in[1];   // [32, 64, 32]
  float* out = (float*)d_out;               // [64, 64, 64]

  float* pred  = (float*)d_ws;                          // 128 MiB
  float* Vg    = pred + (size_t)BS * NUM_IN * OUT_DIM;  // 1 MiB
  float* Spart = Vg + (size_t)BS * 64 * OUT_DIM;        // NSPLIT MiB

  pred_gemm_kernel<<<dim3(NUM_SHARED, BS), 256, 0, stream>>>(x, W, pred);
  zero_v_kernel<<<BS, 256, 0, stream>>>(Vg);
  for (int it = 0; it < ROUTE_NUM; ++it) {
    route_pass_kernel<<<dim3(NSPLIT, BS), 256, 0, stream>>>(pred, Vg, Spart);
    squash_kernel<<<BS, 64, 0, stream>>>(Spart, Vg, out, it == ROUTE_NUM - 1 ? 1 : 0);
  }
}